// FourierEnhancedBlock_85478439125306
// MI455X (gfx1250) — compile-verified
//
#include <hip/hip_runtime.h>
#include <math.h>

// FourierEnhancedBlock on gfx1250: 3 fp32-WMMA GEMM stages replacing rfft->mix->irfft.
// B=32 H=16 L=2048 D=64 M=64. Memory-bound (~545MB HBM => ~23us floor @23.3TB/s).
// Double-buffered GLOBAL_LOAD_ASYNC_TO_LDS_B128 staging (ASYNCcnt) overlaps the
// HBM stream with the v_wmma_f32_16x16x4_f32 pipeline.

typedef float v2f __attribute__((ext_vector_type(2)));
typedef float v4f __attribute__((ext_vector_type(4)));
typedef float v8f __attribute__((ext_vector_type(8)));
typedef int   v4i __attribute__((vector_size(16)));   // matches builtin param type

#define Bsz 32
#define Hn  16
#define Ln  2048
#define Dn  64
#define Mn  64
#define TWO_PI 6.283185307179586f

#if __has_builtin(__builtin_amdgcn_global_load_async_to_lds_b128)
#define FE_ASYNC 1
#else
#define FE_ASYNC 0
#endif

typedef __attribute__((address_space(1))) v4i* gas_v4i;  // global int4*
typedef __attribute__((address_space(3))) v4i* las_v4i;  // LDS int4*

// 16B global->LDS copy: async DMA path on gfx1250 (no VGPR round trip), else sync.
static __device__ __forceinline__ void async_copy16(const float* g, float* l) {
#if FE_ASYNC
  __builtin_amdgcn_global_load_async_to_lds_b128((gas_v4i)g, (las_v4i)l, 0, 0);
#else
  *(v4f*)l = *(const v4f*)g;
#endif
}

template <int N>
static __device__ __forceinline__ void wait_async() {
#if FE_ASYNC
#if __has_builtin(__builtin_amdgcn_s_wait_asynccnt)
  __builtin_amdgcn_s_wait_asynccnt(N);
#else
  asm volatile("s_wait_asynccnt %0" ::"i"(N) : "memory");
#endif
#endif
}

// V_WMMA_F32_16X16X4_F32: D(16x16,f32) = A(16x4,f32) x B(4x16,f32) + C
// A per lane: v2f, lanes0-15 rows M=lane K={kb,kb+1}; lanes16-31 rows M=lane-16 K={kb+2,kb+3}
// B per lane: v2f, VGPR0 rows K={kb | kb+2}, VGPR1 rows K={kb+1 | kb+3}, N=lane%16
// C/D: v8f, VGPR v -> (M=v, lanes0-15) / (M=v+8, lanes16-31)
static __device__ __forceinline__ v8f wmma_f32(v2f a, v2f b, v8f c) {
  return __builtin_amdgcn_wmma_f32_16x16x4_f32(false, a, false, b, (short)0, c, false, false);
}

// ---------------------------------------------------------------- tables ----
// fwd_cos[m][l] = cos(2pi*modes[m]*l/L), fwd_msin = -sin(...)   (X = sum x e^{-i th})
// invB[k][l]    = c_k cos(2pi k l/L)  (k<64),  invB[64+k][l] = -c_k sin(...)
//                 c_0 = 1/L, c_k = 2/L  (row 64 all zeros: Im(bin0) ignored by irfft)
__global__ void __launch_bounds__(256) fe_build_tables(
    const int* __restrict__ modes, float* __restrict__ fwd_cos,
    float* __restrict__ fwd_msin, float* __restrict__ invB) {
  int idx = blockIdx.x * blockDim.x + threadIdx.x;
  const int FWD = Mn * Ln;        // 131072
  const int INV = 2 * Mn * Ln;    // 262144
  if (idx < FWD) {
    int m = idx >> 11;
    int l = idx & (Ln - 1);
    int t = (modes[m] * l) & (Ln - 1);   // exact phase reduction, power-of-two L
    float th = (float)t * (TWO_PI / (float)Ln);
    float s, c;
    __sincosf(th, &s, &c);
    fwd_cos[idx] = c;
    fwd_msin[idx] = -s;
  } else if (idx < FWD + INV) {
    int j = idx - FWD;
    int row = j >> 11;
    int l = j & (Ln - 1);
    int k = row & (Mn - 1);
    int t = (k * l) & (Ln - 1);
    float th = (float)t * (TWO_PI / (float)Ln);
    float s, c;
    __sincosf(th, &s, &c);
    float ck = (k == 0) ? (1.0f / (float)Ln) : (2.0f / (float)Ln);
    invB[j] = (row < Mn) ? (ck * c) : (-ck * s);
  }
}

// ---------------------------------------------------------------- stage 1 ----
// Per (b,h): selT[m][d] = sum_l tw[m][l] * x[b,h,l,d].  64x64 out, K=2048, x2 (re/im).
// 8 waves, each owns tiles (mt,dt)=(p>>2,p&3) for p=wave and p=wave+8 (shares B operand).
// Double-buffered async staging: 6 b128 ops/thread/chunk, wait asynccnt<=6.
#define S1_CH    32
#define XS_STRIDE 72   // 2*72 mod 64 = 16 -> half-wave B rows on disjoint banks
#define CS_STRIDE 36   // mult of 4 (16B-aligned b128), 8B-aligned A reads, conflict-free

__global__ void __launch_bounds__(256) fe_stage1_dft(
    const float* __restrict__ x, const float* __restrict__ fwd_cos,
    const float* __restrict__ fwd_msin, float* __restrict__ sel_r,
    float* __restrict__ sel_i) {
  __shared__ float xs[2][S1_CH * XS_STRIDE];
  __shared__ float cs[2][Mn * CS_STRIDE];
  __shared__ float ss[2][Mn * CS_STRIDE];
  const int bh = blockIdx.x;               // b*H + h
  const int b = bh >> 4, h = bh & 15;
  const int tid = threadIdx.x;
  const int wave = tid >> 5, lane = tid & 31;
  const int half = lane >> 4, ln = lane & 15;
  const int mt0 = wave >> 2, dt = wave & 3;
  const int mt1 = mt0 + 2;

  v8f acc_r0 = {}, acc_i0 = {}, acc_r1 = {}, acc_i1 = {};
  const float* xg = x + (size_t)bh * Ln * Dn;

  auto issue_chunk = [&](int l0, int buf) {
#pragma unroll
    for (int i = 0; i < 2; ++i) {          // x chunk: 32x64 floats
      int f = tid + i * 256;
      int row = f >> 4, c4 = (f & 15) << 2;
      async_copy16(xg + (size_t)(l0 + row) * Dn + c4, &xs[buf][row * XS_STRIDE + c4]);
    }
#pragma unroll
    for (int i = 0; i < 2; ++i) {          // twiddle chunks: 64x32 each
      int f = tid + i * 256;
      int m = f >> 3, c4 = (f & 7) << 2;
      async_copy16(fwd_cos + m * Ln + l0 + c4, &cs[buf][m * CS_STRIDE + c4]);
      async_copy16(fwd_msin + m * Ln + l0 + c4, &ss[buf][m * CS_STRIDE + c4]);
    }
  };

  issue_chunk(0, 0);
  const int NCH = Ln / S1_CH;              // 64 chunks
  for (int c = 0; c < NCH; ++c) {
    int buf = c & 1;
    if (c + 1 < NCH) {
      issue_chunk((c + 1) * S1_CH, buf ^ 1);  // overlap next chunk with compute
      wait_async<6>();                        // chunk c's 6 ops have landed (in-order)
    } else {
      wait_async<0>();
    }
    __syncthreads();

    const float* xb = xs[buf];
    const float* cb = cs[buf];
    const float* sb = ss[buf];
#pragma unroll
    for (int kb = 0; kb < S1_CH; kb += 4) {
      int ko = kb + half * 2;
      v2f a_c0 = *(const v2f*)(&cb[(mt0 * 16 + ln) * CS_STRIDE + ko]);
      v2f a_s0 = *(const v2f*)(&sb[(mt0 * 16 + ln) * CS_STRIDE + ko]);
      v2f a_c1 = *(const v2f*)(&cb[(mt1 * 16 + ln) * CS_STRIDE + ko]);
      v2f a_s1 = *(const v2f*)(&sb[(mt1 * 16 + ln) * CS_STRIDE + ko]);
      v2f bx;
      bx.x = xb[ko * XS_STRIDE + dt * 16 + ln];
      bx.y = xb[(ko + 1) * XS_STRIDE + dt * 16 + ln];
      acc_r0 = wmma_f32(a_c0, bx, acc_r0);
      acc_i0 = wmma_f32(a_s0, bx, acc_i0);
      acc_r1 = wmma_f32(a_c1, bx, acc_r1);
      acc_i1 = wmma_f32(a_s1, bx, acc_i1);
    }
    __syncthreads();                       // protect buf^1 before next async issue
  }

  // sel layout [h][m][b][d] so each (h,m) slice is a contiguous 32x64 GEMM operand
#pragma unroll
  for (int v = 0; v < 8; ++v) {
    int d = dt * 16 + ln;
    int m0 = mt0 * 16 + half * 8 + v;
    int m1 = mt1 * 16 + half * 8 + v;
    size_t o0 = ((size_t)(h * Mn + m0) * Bsz + b) * Dn + d;
    size_t o1 = ((size_t)(h * Mn + m1) * Bsz + b) * Dn + d;
    sel_r[o0] = acc_r0[v];
    sel_i[o0] = acc_i0[v];
    sel_r[o1] = acc_r1[v];
    sel_i[o1] = acc_i1[v];
  }
}

// ---------------------------------------------------------------- stage 2 ----
// Per (h,m): Filt = Sel * W (complex), Sel [32(b) x 64(i)], W [64(i) x 64(j)].
// fr = Sr*Wr - Si*Wi (two accums, VALU subtract at end: f32 WMMA has no A-negate),
// fi = Sr*Wi + Si*Wr (chained accumulator).
#define SR_STRIDE 68
#define W_STRIDE  72

__global__ void __launch_bounds__(256) fe_stage2_mix(
    const float* __restrict__ sel_r, const float* __restrict__ sel_i,
    const float* __restrict__ w_real, const float* __restrict__ w_imag,
    float* __restrict__ Fv) {
  __shared__ float Sr[Bsz * SR_STRIDE];
  __shared__ float Si[Bsz * SR_STRIDE];
  __shared__ float Wr[Dn * W_STRIDE];
  __shared__ float Wi[Dn * W_STRIDE];
  const int blk = blockIdx.x;              // h*Mn + m
  const int h = blk >> 6, m = blk & 63;
  const int tid = threadIdx.x;
  const int wave = tid >> 5, lane = tid & 31;
  const int half = lane >> 4, ln = lane & 15;
  const int bt = wave >> 2, jt = wave & 3;

  const float* srg = sel_r + (size_t)blk * (Bsz * Dn);
  const float* sig = sel_i + (size_t)blk * (Bsz * Dn);
#pragma unroll
  for (int i = 0; i < 2; ++i) {
    int f = tid + i * 256;
    int row = f >> 4, c4 = (f & 15) << 2;
    async_copy16(srg + row * Dn + c4, &Sr[row * SR_STRIDE + c4]);
    async_copy16(sig + row * Dn + c4, &Si[row * SR_STRIDE + c4]);
  }
  wait_async<0>();
#pragma unroll
  for (int s = 0; s < 16; ++s) {           // gather w[h,i,j,m] (L2 resident, 33MB)
    int e = tid + s * 256;
    int i_ = e >> 6, j = e & 63;
    size_t go = ((size_t)(h * Dn + i_) * Dn + j) * Mn + m;
    Wr[i_ * W_STRIDE + j] = w_real[go];
    Wi[i_ * W_STRIDE + j] = w_imag[go];
  }
  __syncthreads();

  v8f acc_rr = {}, acc_ii = {}, acc_im = {};
#pragma unroll
  for (int kb = 0; kb < Dn; kb += 4) {
    int ko = kb + half * 2;
    v2f a_r = *(const v2f*)(&Sr[(bt * 16 + ln) * SR_STRIDE + ko]);
    v2f a_i = *(const v2f*)(&Si[(bt * 16 + ln) * SR_STRIDE + ko]);
    v2f b_r, b_i;
    b_r.x = Wr[ko * W_STRIDE + jt * 16 + ln];
    b_r.y = Wr[(ko + 1) * W_STRIDE + jt * 16 + ln];
    b_i.x = Wi[ko * W_STRIDE + jt * 16 + ln];
    b_i.y = Wi[(ko + 1) * W_STRIDE + jt * 16 + ln];
    acc_rr = wmma_f32(a_r, b_r, acc_rr);
    acc_ii = wmma_f32(a_i, b_i, acc_ii);
    acc_im = wmma_f32(a_r, b_i, acc_im);
    acc_im = wmma_f32(a_i, b_r, acc_im);
  }

  // F[b][h][j][128]: Re at [.][m], Im at [.][64+m]  -> stage3 A operand
#pragma unroll
  for (int v = 0; v < 8; ++v) {
    int bb = bt * 16 + half * 8 + v;
    int j = jt * 16 + ln;
    size_t base = ((size_t)(bb * Hn + h) * Dn + j) * 128;
    Fv[base + m] = acc_rr[v] - acc_ii[v];
    Fv[base + Mn + m] = acc_im[v];
  }
}

// ---------------------------------------------------------------- stage 3 ----
// Per (b,h): out[d][l] = sum_{k<128} F[d][k] * invB[k][l].  M=64, N=2048, K=128.
// Fs loaded async once; Bs double-buffered (8 b128 ops/thread/chunk, wait<=8).
#define FS_STRIDE 132
#define BS_STRIDE 72

__global__ void __launch_bounds__(256) fe_stage3_idft(
    const float* __restrict__ Fv, const float* __restrict__ invB,
    float* __restrict__ out) {
  __shared__ float Fs[Dn * FS_STRIDE];       // 64 x 128 (+pad)
  __shared__ float Bs[2][128 * BS_STRIDE];   // 128 x 64 chunks (+pad), double buffered
  const int bh = blockIdx.x;
  const int tid = threadIdx.x;
  const int wave = tid >> 5, lane = tid & 31;
  const int half = lane >> 4, ln = lane & 15;
  const int mt0 = wave >> 2, nt = wave & 3;
  const int mt1 = mt0 + 2;

  const float* fg = Fv + (size_t)bh * (Dn * 128);
#pragma unroll
  for (int s = 0; s < 8; ++s) {              // Fs: 64x128 floats, async
    int f = tid + s * 256;
    int row = f >> 5, c4 = (f & 31) << 2;
    async_copy16(fg + row * 128 + c4, &Fs[row * FS_STRIDE + c4]);
  }

  auto issue_b = [&](int nc, int buf) {
#pragma unroll
    for (int s = 0; s < 8; ++s) {            // Bs chunk: 128x64 floats
      int f = tid + s * 256;
      int row = f >> 4, c4 = (f & 15) << 2;
      async_copy16(invB + row * Ln + nc + c4, &Bs[buf][row * BS_STRIDE + c4]);
    }
  };

  float* og = out + (size_t)bh * Dn * Ln;
  issue_b(0, 0);

  const int NCH = Ln / 64;                   // 32 chunks
  for (int c = 0; c < NCH; ++c) {
    int buf = c & 1;
    if (c + 1 < NCH) {
      issue_b((c + 1) * 64, buf ^ 1);        // overlap next chunk with compute
      wait_async<8>();                       // Fs + chunk c landed (in-order)
    } else {
      wait_async<0>();
    }
    __syncthreads();

    const float* bb = Bs[buf];
    v8f acc0 = {}, acc1 = {};
#pragma unroll 8
    for (int kb = 0; kb < 128; kb += 4) {
      int ko = kb + half * 2;
      v2f bv;
      bv.x = bb[ko * BS_STRIDE + nt * 16 + ln];
      bv.y = bb[(ko + 1) * BS_STRIDE + nt * 16 + ln];
      v2f a0 = *(const v2f*)(&Fs[(mt0 * 16 + ln) * FS_STRIDE + ko]);
      v2f a1 = *(const v2f*)(&Fs[(mt1 * 16 + ln) * FS_STRIDE + ko]);
      acc0 = wmma_f32(a0, bv, acc0);
      acc1 = wmma_f32(a1, bv, acc1);
    }

#pragma unroll
    for (int v = 0; v < 8; ++v) {
      int l = c * 64 + nt * 16 + ln;
      int d0 = mt0 * 16 + half * 8 + v;
      int d1 = mt1 * 16 + half * 8 + v;
      og[(size_t)d0 * Ln + l] = acc0[v];     // lanes 0-15 / 16-31 -> 64B coalesced runs
      og[(size_t)d1 * Ln + l] = acc1[v];
    }
    __syncthreads();                         // protect buf^1 before next async issue
  }
}

// ---------------------------------------------------------------- launch ----
extern "C" void kernel_launch(void* const* d_in, const int* in_sizes, int n_in,
                              void* d_out, int out_size, void* d_ws, size_t ws_size,
                              hipStream_t stream) {
  (void)in_sizes; (void)n_in; (void)out_size; (void)ws_size;
  const float* x      = (const float*)d_in[0];
  const float* w_real = (const float*)d_in[1];
  const float* w_imag = (const float*)d_in[2];
  const int*   modes  = (const int*)d_in[3];
  float* out = (float*)d_out;
  float* ws  = (float*)d_ws;

  // workspace layout (floats), total ~35.7 MB -> L2 resident
  float* fwd_cos  = ws;                                 // M*L   = 131072
  float* fwd_msin = fwd_cos + Mn * Ln;                  // M*L   = 131072
  float* invB     = fwd_msin + Mn * Ln;                 // 2M*L  = 262144
  float* sel_r    = invB + 2 * Mn * Ln;                 // H*M*B*D = 2097152
  float* sel_i    = sel_r + (size_t)Hn * Mn * Bsz * Dn;
  float* Fv       = sel_i + (size_t)Hn * Mn * Bsz * Dn; // B*H*D*128 = 4194304

  int tab_elems = Mn * Ln + 2 * Mn * Ln;                // 393216
  fe_build_tables<<<(tab_elems + 255) / 256, 256, 0, stream>>>(modes, fwd_cos, fwd_msin, invB);
  fe_stage1_dft<<<Bsz * Hn, 256, 0, stream>>>(x, fwd_cos, fwd_msin, sel_r, sel_i);
  fe_stage2_mix<<<Hn * Mn, 256, 0, stream>>>(sel_r, sel_i, w_real, w_imag, Fv);
  fe_stage3_idft<<<Bsz * Hn, 256, 0, stream>>>(Fv, invB, out);
}